// SE3InvariantGraphEncoder_51058571215446
// MI455X (gfx1250) — compile-verified
//
#include <hip/hip_runtime.h>
#include <math.h>

typedef float v2f  __attribute__((ext_vector_type(2)));
typedef float v8f  __attribute__((ext_vector_type(8)));
typedef unsigned int u32x4 __attribute__((ext_vector_type(4)));
typedef int   i32x4 __attribute__((ext_vector_type(4)));
typedef int   i32x8 __attribute__((ext_vector_type(8)));

#define HID_    64
#define NUMEL_  1024
#define TILE_E  64
// TP_NORM * DOT_NORM * SCALE = (1/8)*(1/4)*(1/2) = 1/64
#define TP_NORM_F     0.125f
#define LOGIT_SCALE_F 0.015625f

#if __has_builtin(__builtin_amdgcn_tensor_load_to_lds) && __has_builtin(__builtin_amdgcn_s_wait_tensorcnt)
#define HAVE_TDM 1
#else
#define HAVE_TDM 0
#endif

__device__ __forceinline__ unsigned fkey(float f){
  unsigned u = __float_as_uint(f);
  return (u & 0x80000000u) ? ~u : (u | 0x80000000u);
}
__device__ __forceinline__ float funkey(unsigned k){
  unsigned u = (k & 0x80000000u) ? (k & 0x7FFFFFFFu) : ~k;
  return __uint_as_float(u);
}
__device__ __forceinline__ void lds_atomic_add(float* p, float v){
  __hip_atomic_fetch_add(p, v, __ATOMIC_RELAXED, __HIP_MEMORY_SCOPE_WORKGROUP);
}
__device__ __forceinline__ void g_atomic_add(float* p, float v){
  __hip_atomic_fetch_add(p, v, __ATOMIC_RELAXED, __HIP_MEMORY_SCOPE_AGENT);
}

// ---- B-slab staging: 64x1024 f32 (256KB) global -> LDS, via TDM when available
__device__ __forceinline__ void load_b_slab(float* Bsh, const float* Bg, int tid, int wave){
  __syncthreads();            // all waves done reading previous slab
#if HAVE_TDM
  if (wave == 0){
    unsigned long long ga = (unsigned long long)(uintptr_t)Bg;
    unsigned lds = (unsigned)(uintptr_t)Bsh;    // low 32 bits = LDS offset
    u32x4 g0;
    g0.x = 1u;                                       // count=1, user descriptor
    g0.y = lds;                                      // lds_addr [63:32]
    g0.z = (unsigned)(ga & 0xFFFFFFFFu);             // global_addr lo
    g0.w = (unsigned)((ga >> 32) & 0x01FFFFFFu) | (2u << 30); // addr hi | type=2
    i32x8 g1;
    g1[0] = 0x00020000;        // data_size=2 (4B), no multicast/pad/iterate
    g1[1] = (int)0x04000000u;  // tensor_dim0 = 1024 (lo16 @ bits 63:48)
    g1[2] = (int)0x00400000u;  // tensor_dim0 hi = 0 | tensor_dim1 = 64 (lo16)
    g1[3] = (int)0x04000000u;  // tensor_dim1 hi = 0 | tile_dim0 = 1024
    g1[4] = 64;                // tile_dim1 = 64, tile_dim2 = 0
    g1[5] = 1024;              // tensor_dim0_stride lo32
    g1[6] = 0;                 // stride hi / dim1_stride lo
    g1[7] = 0;
    i32x4 gz4 = {0,0,0,0};
    i32x8 gz8 = {0,0,0,0,0,0,0,0};
    __builtin_amdgcn_tensor_load_to_lds(g0, g1, gz4, gz4, gz8, 0);
    __builtin_amdgcn_s_wait_tensorcnt(0);
  }
  __syncthreads();
#else
  for (int i = tid; i < (HID_*NUMEL_)/4; i += 256)
    ((float4*)Bsh)[i] = ((const float4*)Bg)[i];
  __syncthreads();
#endif
}

// ---- prep kernels -----------------------------------------------------------
__global__ void kzero(unsigned* __restrict__ mkey, float* __restrict__ den,
                      float* __restrict__ agg, int Ntot){
  int i = blockIdx.x*256 + threadIdx.x;
  if (i < Ntot*4){ mkey[i] = 0u; den[i] = 0.f; }
  if (i < Ntot*16) agg[i] = 0.f;
}

// permute W2 [h2][p*16+k] into WMMA B-fragment order:
// frag[((j*16+t)*32 + lane)*2 + i] = W2[j][ (t*4 + 2*(lane>>4) + i)*16 + (lane&15) ]
__global__ void kperm(const float* __restrict__ kW2, const float* __restrict__ vW2,
                      float* __restrict__ Bk, float* __restrict__ Bv){
  int o = blockIdx.x*256 + threadIdx.x;
  if (o >= HID_*NUMEL_) return;
  int i = o & 1;
  int l = (o >> 1) & 31;
  int t = (o >> 6) & 15;
  int j = o >> 10;
  int p = t*4 + ((l >> 4) << 1) + i;
  int k = l & 15;
  int src = j*1024 + p*16 + k;
  Bk[o] = kW2[src];
  Bv[o] = vW2[src];
}

// qd[n, h*4+j] = sum_i (x@Wq)[n,h*4+i] * Wdot[i,j]
__global__ void kqd(const float* __restrict__ nf, const float* __restrict__ Wq,
                    const float* __restrict__ Wdot, float* __restrict__ qd, int Ntot){
  int n = blockIdx.x*256 + threadIdx.x;
  if (n >= Ntot) return;
  float x[16], q[16];
  #pragma unroll
  for (int c=0;c<16;c++) x[c] = nf[n*16+c];
  #pragma unroll
  for (int c2=0;c2<16;c2++){
    float s = 0.f;
    #pragma unroll
    for (int c=0;c<16;c++) s += x[c]*Wq[c*16+c2];
    q[c2] = s;
  }
  #pragma unroll
  for (int h=0;h<4;h++)
    #pragma unroll
    for (int j=0;j<4;j++){
      float s = 0.f;
      #pragma unroll
      for (int i=0;i<4;i++) s += q[h*4+i]*Wdot[i*4+j];
      qd[n*16 + h*4 + j] = s;
    }
}

// ---- main edge kernel -------------------------------------------------------
__global__ __launch_bounds__(256) void k2_edge(
    const float* __restrict__ nf,  const float* __restrict__ esh,
    const float* __restrict__ emb, const float* __restrict__ elen,
    const float* __restrict__ kW1, const float* __restrict__ kb1,
    const float* __restrict__ vW1, const float* __restrict__ vb1,
    const float* __restrict__ kb2, const float* __restrict__ vb2,
    const int*   __restrict__ eidx,
    const float* __restrict__ Bk,  const float* __restrict__ Bv,
    const float* __restrict__ qd,
    float* __restrict__ logits, float* __restrict__ vedge,
    unsigned* __restrict__ mkey, int Etot)
{
  extern __shared__ float smem[];
  float* Bsh   = smem;                    // 64*1024
  float* outer = Bsh + HID_*NUMEL_;       // 64 edges x 68 (padded)
  float* hidb  = outer + 64*68;           // 2 x 64 x 68
  float* kvt   = hidb + 2*64*68;          // 16 x 17
  int*   dsth  = (int*)(kvt + 16*17);     // 64
  float* cutf  = (float*)(dsth + 64);     // 64

  const int tid  = (int)threadIdx.x;
  const int lane = tid & 31;
  const int wave = tid >> 5;
  const int e0   = (int)blockIdx.x * TILE_E;
  const v8f zero8 = {0,0,0,0,0,0,0,0};

  // ---- stage outer[e][p] = x_src[e][c]*sh[e][s], dst, cutoff ----
  {
    int e = tid >> 2, q = tid & 3;
    int ge = e0 + e;
    bool valid = ge < Etot;
    int gs = valid ? eidx[ge] : 0;
    float shv[4];
    #pragma unroll
    for (int s=0;s<4;s++) shv[s] = valid ? esh[ge*4+s] : 0.f;
    #pragma unroll
    for (int i=0;i<4;i++){
      float xv = valid ? nf[gs*16 + q*4 + i] : 0.f;
      #pragma unroll
      for (int s=0;s<4;s++) outer[e*68 + (q*4+i)*4 + s] = xv * shv[s];
    }
    if (q == 0){
      dsth[e] = valid ? eidx[Etot + ge] : 0;
      float L = valid ? elen[ge] : 0.f;
      cutf[e] = 1.f / (1.f + __expf(-10.f*(1.f - L*0.2f)));
    }
  }
  // ---- radial MLP layer 1 + silu for both branches ----
  for (int it=0; it<32; it++){
    int o  = tid + (it << 8);
    int br = o >> 12, rem = o & 4095, e = rem >> 6, h2 = rem & 63;
    int ge = e0 + e;
    bool valid = ge < Etot;
    const float* W1 = br ? vW1 : kW1;
    const float* b1 = br ? vb1 : kb1;
    float acc = b1[h2];
    if (valid){
      #pragma unroll
      for (int b=0;b<16;b++) acc += emb[ge*16+b] * W1[b*64 + h2];
    }
    float hv = acc / (1.f + __expf(-acc));
    hidb[br*(64*68) + e*68 + h2] = valid ? hv : 0.f;
  }
  __syncthreads();

  const int m      = lane & 15;
  const int khalf  = (lane >> 4) << 1;
  const int rowoff = (lane >> 4) << 3;

  for (int br=0; br<2; br++){
    load_b_slab(Bsh, br ? Bv : Bk, tid, wave);
    const float* b2    = br ? vb2 : kb2;
    const float* hslab = hidb + br*(64*68);

    for (int et=0; et<4; et++){
      for (int z=tid; z<16*17; z+=256) kvt[z] = 0.f;
      __syncthreads();

      const float* Arow = &outer[(et*16 + m)*68];
      v8f acc[8];
      #pragma unroll
      for (int j=0;j<8;j++) acc[j] = zero8;

      for (int t=0;t<16;t++){
        v2f a = *(const v2f*)(Arow + t*4 + khalf);
        #pragma unroll
        for (int j=0;j<8;j++){
          int nt = wave*8 + j;
          v2f b = *(const v2f*)(Bsh + ((nt*16 + t)*32 + lane)*2);
          acc[j] = __builtin_amdgcn_wmma_f32_16x16x4_f32(
                     false, a, false, b, (short)0, acc[j], false, false);
        }
      }
      // bias tile: Outer @ b2[64x16], wave 0 only (uniform branch, EXEC all-1s)
      v8f part = zero8;
      if (wave == 0){
        for (int t=0;t<16;t++){
          v2f a = *(const v2f*)(Arow + t*4 + khalf);
          int p0 = t*4 + khalf;
          v2f b; b.x = b2[p0*16 + m]; b.y = b2[(p0+1)*16 + m];
          part = __builtin_amdgcn_wmma_f32_16x16x4_f32(
                   false, a, false, b, (short)0, part, false, false);
        }
      }
      // contract with hidden activations: kv[e,k] = sum_h2 hid[e,h2]*T[e,h2,k]
      #pragma unroll
      for (int j=0;j<8;j++){
        int h2 = wave*8 + j;
        #pragma unroll
        for (int r=0;r<8;r++){
          float hv = hslab[(et*16 + r + rowoff)*68 + h2];
          part[r] += hv * acc[j][r];
        }
      }
      #pragma unroll
      for (int r=0;r<8;r++)
        lds_atomic_add(&kvt[(r + rowoff)*17 + m], part[r]);
      __syncthreads();

      if (br == 0){
        if (tid < 64){
          int eL = tid >> 2, h = tid & 3;
          int ge = e0 + et*16 + eL;
          if (ge < Etot){
            int d = dsth[et*16 + eL];
            float dot = 0.f;
            #pragma unroll
            for (int j=0;j<4;j++)
              dot += qd[d*16 + h*4 + j] * kvt[eL*17 + h*4 + j];
            float lg = dot * LOGIT_SCALE_F * cutf[et*16 + eL];
            logits[ge*4 + h] = lg;
            atomicMax(&mkey[d*4 + h], fkey(lg));
          }
        }
      } else {
        int eL = tid >> 4, c = tid & 15;
        int ge = e0 + et*16 + eL;
        if (ge < Etot) vedge[ge*16 + c] = kvt[eL*17 + c] * TP_NORM_F;
      }
      __syncthreads();
    }
  }
}

// ---- pass 2: scatter softmax numerators + weighted value aggregation --------
__global__ void k3_attn(const float* __restrict__ logits, const float* __restrict__ vedge,
                        const unsigned* __restrict__ mkey, const int* __restrict__ eidx,
                        float* __restrict__ den, float* __restrict__ agg, int Etot){
  int e = blockIdx.x*256 + threadIdx.x;
  if (e >= Etot) return;
  int d = eidx[Etot + e];
  float ex[4];
  #pragma unroll
  for (int h=0;h<4;h++){
    float mm = funkey(mkey[d*4+h]);
    float v  = __expf(logits[e*4+h] - mm);
    ex[h] = v;
    g_atomic_add(&den[d*4+h], v);
  }
  #pragma unroll
  for (int c=0;c<16;c++)
    g_atomic_add(&agg[d*16+c], ex[c>>2]*vedge[e*16+c]);
}

// ---- pass 3: per-node output projection + FFN (WMMA) ------------------------
__global__ __launch_bounds__(32) void k4_node(
    const float* __restrict__ nf,   const float* __restrict__ agg,
    const float* __restrict__ den,  const float* __restrict__ Wout,
    const float* __restrict__ Wf1,  const float* __restrict__ Wf2,
    float* __restrict__ out, int Ntot)
{
  __shared__ float t1[16*18];
  __shared__ float t2[16*34];
  const int lane  = (int)threadIdx.x;
  const int m     = lane & 15;
  const int half  = lane >> 4;
  const int khalf = half << 1;
  const int n0    = (int)blockIdx.x * 16;
  const v8f zero8 = {0,0,0,0,0,0,0,0};

  // attn_agg/den @ W_out
  v8f acc = zero8;
  #pragma unroll
  for (int t=0;t<4;t++){
    int kc   = t*4 + khalf;
    int nrow = n0 + m; if (nrow >= Ntot) nrow = Ntot - 1;
    float dv  = den[nrow*4 + t];
    float inv = (dv != 0.f) ? 1.f/dv : 0.f;
    v2f a; a.x = agg[nrow*16 + kc]*inv; a.y = agg[nrow*16 + kc + 1]*inv;
    v2f b; b.x = Wout[kc*16 + m];       b.y = Wout[(kc+1)*16 + m];
    acc = __builtin_amdgcn_wmma_f32_16x16x4_f32(false, a, false, b, (short)0, acc, false, false);
  }
  v8f ao;
  #pragma unroll
  for (int r=0;r<8;r++){
    int row = n0 + r + half*8; if (row >= Ntot) row = Ntot - 1;
    ao[r] = nf[row*16 + m] + acc[r];
    t1[(r + half*8)*18 + m] = ao[r];
  }
  __syncthreads();
  // attn_out @ W_f1 (16x32)
  v8f f0 = zero8, f1v = zero8;
  #pragma unroll
  for (int t=0;t<4;t++){
    v2f a = *(const v2f*)(&t1[m*18 + t*4 + khalf]);
    int p0 = t*4 + khalf;
    { v2f b; b.x = Wf1[p0*32 + m];      b.y = Wf1[(p0+1)*32 + m];
      f0  = __builtin_amdgcn_wmma_f32_16x16x4_f32(false, a, false, b, (short)0, f0,  false, false); }
    { v2f b; b.x = Wf1[p0*32 + 16 + m]; b.y = Wf1[(p0+1)*32 + 16 + m];
      f1v = __builtin_amdgcn_wmma_f32_16x16x4_f32(false, a, false, b, (short)0, f1v, false, false); }
  }
  // norm-act: x * sigmoid(|x|)
  #pragma unroll
  for (int r=0;r<8;r++){
    float x0 = f0[r], x1 = f1v[r];
    t2[(r+half*8)*34 + m]      = x0 / (1.f + __expf(-fabsf(x0)));
    t2[(r+half*8)*34 + 16 + m] = x1 / (1.f + __expf(-fabsf(x1)));
  }
  __syncthreads();
  // g @ W_f2 (32x16), K=32
  v8f ff = zero8;
  #pragma unroll
  for (int t=0;t<8;t++){
    v2f a = *(const v2f*)(&t2[m*34 + t*4 + khalf]);
    int p0 = t*4 + khalf;
    v2f b; b.x = Wf2[p0*16 + m]; b.y = Wf2[(p0+1)*16 + m];
    ff = __builtin_amdgcn_wmma_f32_16x16x4_f32(false, a, false, b, (short)0, ff, false, false);
  }
  #pragma unroll
  for (int r=0;r<8;r++){
    int row = n0 + r + half*8;
    if (row < Ntot) out[row*16 + m] = ao[r] + ff[r];
  }
}

// ---- host launcher ----------------------------------------------------------
extern "C" void kernel_launch(void* const* d_in, const int* in_sizes, int n_in,
                              void* d_out, int out_size, void* d_ws, size_t ws_size,
                              hipStream_t stream){
  const float* nf   = (const float*)d_in[0];
  const float* esh  = (const float*)d_in[1];
  const float* emb  = (const float*)d_in[2];
  const float* elen = (const float*)d_in[3];
  const float* Wq   = (const float*)d_in[4];
  const float* kW1  = (const float*)d_in[5];
  const float* kb1  = (const float*)d_in[6];
  const float* kW2  = (const float*)d_in[7];
  const float* kb2  = (const float*)d_in[8];
  const float* vW1  = (const float*)d_in[9];
  const float* vb1  = (const float*)d_in[10];
  const float* vW2  = (const float*)d_in[11];
  const float* vb2  = (const float*)d_in[12];
  const float* Wdot = (const float*)d_in[13];
  const float* Wout = (const float*)d_in[14];
  const float* Wf1  = (const float*)d_in[15];
  const float* Wf2  = (const float*)d_in[16];
  const int*   eidx = (const int*)d_in[17];
  float* out = (float*)d_out;

  const int Ntot = in_sizes[0] / 16;
  const int Etot = in_sizes[3];

  float* ws = (float*)d_ws;
  size_t off = 0;
  float* Bk     = ws + off; off += (size_t)HID_*NUMEL_;
  float* Bv     = ws + off; off += (size_t)HID_*NUMEL_;
  float* qd     = ws + off; off += (size_t)Ntot*16;
  float* logit  = ws + off; off += (size_t)Etot*4;
  float* vedge  = ws + off; off += (size_t)Etot*16;
  unsigned* mkey = (unsigned*)(ws + off); off += (size_t)Ntot*4;
  float* den    = ws + off; off += (size_t)Ntot*4;
  float* agg    = ws + off; off += (size_t)Ntot*16;

  kzero<<<(Ntot*16 + 255)/256, 256, 0, stream>>>(mkey, den, agg, Ntot);
  kperm<<<(HID_*NUMEL_ + 255)/256, 256, 0, stream>>>(kW2, vW2, Bk, Bv);
  kqd  <<<(Ntot + 255)/256, 256, 0, stream>>>(nf, Wq, Wdot, qd, Ntot);

  size_t smem = (size_t)(HID_*NUMEL_ + 64*68 + 2*64*68 + 16*17 + 64 + 64) * sizeof(float);
  k2_edge<<<(Etot + TILE_E - 1)/TILE_E, 256, smem, stream>>>(
      nf, esh, emb, elen, kW1, kb1, vW1, vb1, kb2, vb2, eidx,
      Bk, Bv, qd, logit, vedge, mkey, Etot);

  k3_attn<<<(Etot + 255)/256, 256, 0, stream>>>(logit, vedge, mkey, eidx, den, agg, Etot);
  k4_node<<<(Ntot + 15)/16, 32, 0, stream>>>(nf, agg, den, Wout, Wf1, Wf2, out, Ntot);
}